// GPT2Attention_1279900254774
// MI455X (gfx1250) — compile-verified
//
#include <hip/hip_runtime.h>

typedef __attribute__((ext_vector_type(2))) float v2f;
typedef __attribute__((ext_vector_type(8))) float v8f;
typedef __attribute__((ext_vector_type(4))) int v4i;

#define WMMA_F32(a, b, c) \
  __builtin_amdgcn_wmma_f32_16x16x4_f32(false, (a), false, (b), (short)0, (c), false, false)

static constexpr int Bsz = 4, Sq = 2048, Edim = 1024, Hh = 16, Dh = 64;

// ---- CDNA5 async global->LDS path (ASYNCcnt), with sync fallback -----------
#if defined(__has_builtin)
#if __has_builtin(__builtin_amdgcn_global_load_async_to_lds_b128) && \
    __has_builtin(__builtin_amdgcn_global_load_async_to_lds_b32) &&  \
    __has_builtin(__builtin_amdgcn_s_wait_asynccnt)
#define HAVE_ASYNC_LDS 1
#endif
#endif

#ifdef HAVE_ASYNC_LDS
typedef __attribute__((address_space(3))) void* lds_vptr;
typedef __attribute__((address_space(3))) v4i* lds_v4i;
typedef __attribute__((address_space(3))) int* lds_int;
__device__ __forceinline__ void async_b128(const float* g, float* l) {
  // probe-derived signature: (v4i* src, v4i AS3* dst, imm offset, imm cpol)
  __builtin_amdgcn_global_load_async_to_lds_b128((v4i*)g, (lds_v4i)(lds_vptr)l, 0, 0);
}
__device__ __forceinline__ void async_b32(const float* g, float* l) {
  __builtin_amdgcn_global_load_async_to_lds_b32((int*)g, (lds_int)(lds_vptr)l, 0, 0);
}
__device__ __forceinline__ void async_wait0() { __builtin_amdgcn_s_wait_asynccnt(0); }
#endif

__device__ __forceinline__ v8f vzero8() {
  v8f z = {0.f, 0.f, 0.f, 0.f, 0.f, 0.f, 0.f, 0.f};
  return z;
}

// ---------------------------------------------------------------------------
// WMMA GEMM: X[M,1024] @ W[1024,LDN slice] + bias.
// 256 threads (8 waves), 128x128 tile, wave = 32x64 strip (2x4 accumulators).
// K chunked x16 through double-buffered K-major LDS tiles (row pad 132 keeps
// async b128 LDS stores 16B-aligned and fragment reads bank-conflict-free).
// QKV=true scatters to Q/K/V [B,H,S,D]; else dense [M,1024] store.
// ---------------------------------------------------------------------------
template <int LDN, bool QKV>
__global__ __launch_bounds__(256) void gemm_kernel(
    const float* __restrict__ X, const float* __restrict__ W,
    const float* __restrict__ bias, float* __restrict__ O0,
    float* __restrict__ O1, float* __restrict__ O2) {
  constexpr int KC = 16;
  __shared__ float As[2][KC][132];   // K-major: As[k][m]
  __shared__ float Bs[2][KC][132];   // K-major: Bs[k][n]

  const int n0b = blockIdx.x * 128;
  const int m0b = blockIdx.y * 128;
  const int tid = threadIdx.x;
  const int wv = tid >> 5, ln = tid & 31;
  const int row16 = ln & 15, hi = ln >> 4;
  const int m0 = (wv & 3) * 32;      // wave's M offset within tile
  const int n0w = (wv >> 2) * 64;    // wave's N offset within tile

  auto load_tiles = [&](int kc, int buf) {
#ifdef HAVE_ASYNC_LDS
    for (int i = 0; i < 2; ++i) {    // B tile: 512 b128 chunks
      const int idx = tid + i * 256;
      const int r = idx >> 5, c4 = (idx & 31) * 4;
      async_b128(W + (size_t)(kc + r) * LDN + n0b + c4, &Bs[buf][r][c4]);
    }
    for (int i = 0; i < 8; ++i) {    // A tile transposed: per-element b32
      const int idx = tid + i * 256;
      const int m = idx >> 4, k = idx & 15;
      async_b32(X + (size_t)(m0b + m) * Edim + kc + k, &As[buf][k][m]);
    }
#else
    for (int i = 0; i < 2; ++i) {
      const int idx = tid + i * 256;
      const int r = idx >> 5, c4 = (idx & 31) * 4;
      const float4 v = *(const float4*)(W + (size_t)(kc + r) * LDN + n0b + c4);
      Bs[buf][r][c4 + 0] = v.x; Bs[buf][r][c4 + 1] = v.y;
      Bs[buf][r][c4 + 2] = v.z; Bs[buf][r][c4 + 3] = v.w;
    }
    for (int i = 0; i < 8; ++i) {
      const int idx = tid + i * 256;
      const int m = idx >> 4, k = idx & 15;
      As[buf][k][m] = X[(size_t)(m0b + m) * Edim + kc + k];
    }
    __builtin_prefetch(W + (size_t)(kc + KC + (tid & 15)) * LDN + n0b, 0, 0);
#endif
  };

  v8f acc[2][4];
  for (int mi = 0; mi < 2; ++mi)
    for (int t = 0; t < 4; ++t) acc[mi][t] = vzero8();

  load_tiles(0, 0);
#ifdef HAVE_ASYNC_LDS
  async_wait0();
#endif
  __syncthreads();

  int buf = 0;
  for (int kc = 0; kc < Edim; kc += KC) {
    if (kc + KC < Edim) load_tiles(kc + KC, buf ^ 1);  // prefetch next chunk

    for (int ks = 0; ks < KC; ks += 4) {
      const int k0 = ks + hi * 2;
      v2f a[2];
      for (int mi = 0; mi < 2; ++mi) {
        a[mi].x = As[buf][k0 + 0][m0 + mi * 16 + row16];
        a[mi].y = As[buf][k0 + 1][m0 + mi * 16 + row16];
      }
      for (int t = 0; t < 4; ++t) {
        v2f bf;
        bf.x = Bs[buf][k0 + 0][n0w + t * 16 + row16];
        bf.y = Bs[buf][k0 + 1][n0w + t * 16 + row16];
        acc[0][t] = WMMA_F32(a[0], bf, acc[0][t]);
        acc[1][t] = WMMA_F32(a[1], bf, acc[1][t]);
      }
    }
#ifdef HAVE_ASYNC_LDS
    async_wait0();
#endif
    __syncthreads();
    buf ^= 1;
  }

  for (int mi = 0; mi < 2; ++mi) {
    for (int t = 0; t < 4; ++t) {
      const int ncol = n0b + n0w + t * 16 + row16;
      const float bv = bias[ncol];
      for (int i = 0; i < 8; ++i) {
        const int m = m0b + m0 + mi * 16 + hi * 8 + i;
        if constexpr (QKV) {
          const int which = ncol >> 10;
          const int e = ncol & 1023;
          const int h = e >> 6, d = e & 63;
          float* dst = (which == 0) ? O0 : ((which == 1) ? O1 : O2);
          const int bidx = m >> 11, s = m & 2047;
          dst[(((size_t)bidx * Hh + h) * Sq + s) * Dh + d] = acc[mi][t][i] + bv;
        } else {
          O0[(size_t)m * Edim + ncol] = acc[mi][t][i] + bv;
        }
      }
    }
  }
}

// ---------------------------------------------------------------------------
// Causal flash attention: one (b,h, 64-row q-tile) per 128-thread block.
// Double-buffered K/V tiles streamed with async-to-LDS; pad 68 (272B = 17x16)
// keeps b128 LDS targets aligned and K^T fragment reads conflict-free.
// ---------------------------------------------------------------------------
__global__ __launch_bounds__(128) void attn_kernel(
    const float* __restrict__ Qb, const float* __restrict__ Kb,
    const float* __restrict__ Vb, float* __restrict__ Ob) {
  __shared__ float Ks[2][64][68];
  __shared__ float Vs[2][64][68];
  __shared__ float Ps[4][16][68];   // wave-private P staging (C -> A layout)

  const int qt = blockIdx.x;
  const int bh = blockIdx.y;
  const int tid = threadIdx.x, wv = tid >> 5, ln = tid & 31;
  const int row16 = ln & 15, hi = ln >> 4;
  const int q0 = qt * 64;
  const float scale = 0.125f;       // 1/sqrt(64)

  const float* Qh = Qb + (size_t)bh * Sq * Dh;
  const float* Kh = Kb + (size_t)bh * Sq * Dh;
  const float* Vh = Vb + (size_t)bh * Sq * Dh;

  auto load_kv = [&](int kt, int buf) {
    const int k0 = kt * 64;
    for (int i = 0; i < 8; ++i) {
      const int idx = tid + i * 128;
      const int r = idx >> 4, c4 = (idx & 15) * 4;
#ifdef HAVE_ASYNC_LDS
      async_b128(Kh + (size_t)(k0 + r) * Dh + c4, &Ks[buf][r][c4]);
      async_b128(Vh + (size_t)(k0 + r) * Dh + c4, &Vs[buf][r][c4]);
#else
      const float4 kv = *(const float4*)(Kh + (size_t)(k0 + r) * Dh + c4);
      Ks[buf][r][c4 + 0] = kv.x; Ks[buf][r][c4 + 1] = kv.y;
      Ks[buf][r][c4 + 2] = kv.z; Ks[buf][r][c4 + 3] = kv.w;
      const float4 vv = *(const float4*)(Vh + (size_t)(k0 + r) * Dh + c4);
      Vs[buf][r][c4 + 0] = vv.x; Vs[buf][r][c4 + 1] = vv.y;
      Vs[buf][r][c4 + 2] = vv.z; Vs[buf][r][c4 + 3] = vv.w;
#endif
    }
  };

  // Q fragments for this wave's 16 rows, pre-scaled by 1/sqrt(D)
  v2f qa[16];
  {
    const float* qrow = Qh + (size_t)(q0 + wv * 16 + row16) * Dh;
    for (int kk = 0; kk < 16; ++kk) {
      const int d = kk * 4 + hi * 2;
      qa[kk].x = qrow[d + 0] * scale;
      qa[kk].y = qrow[d + 1] * scale;
    }
  }

  v8f o[4];
  float mrow[8], lrow[8];
  for (int t = 0; t < 4; ++t) o[t] = vzero8();
  for (int i = 0; i < 8; ++i) { mrow[i] = -3.0e38f; lrow[i] = 0.f; }

  load_kv(0, 0);
#ifdef HAVE_ASYNC_LDS
  async_wait0();
#endif
  __syncthreads();

  int buf = 0;
  for (int kt = 0; kt <= qt; ++kt) {        // causal: never touch upper tiles
    const int k0 = kt * 64;
    if (kt < qt) load_kv(kt + 1, buf ^ 1);  // stream next K/V tile

    // S = (Q*scale) @ K^T  (16x64 per wave)
    v8f s[4];
    for (int t = 0; t < 4; ++t) s[t] = vzero8();
    for (int kk = 0; kk < 16; ++kk) {
      const int dd = kk * 4 + hi * 2;
      for (int t = 0; t < 4; ++t) {
        v2f bf;
        bf.x = Ks[buf][t * 16 + row16][dd + 0];
        bf.y = Ks[buf][t * 16 + row16][dd + 1];
        s[t] = WMMA_F32(qa[kk], bf, s[t]);
      }
    }

    if (kt == qt) {                          // diagonal tile causal mask
      for (int t = 0; t < 4; ++t) {
        const int key = k0 + t * 16 + row16;
        for (int i = 0; i < 8; ++i) {
          const int q = q0 + wv * 16 + hi * 8 + i;
          if (key > q) s[t][i] = -3.0e38f;
        }
      }
    }

    // online softmax (rows live in 16-lane halves of the wave)
    float alpha[8];
    for (int i = 0; i < 8; ++i) {
      float mx = fmaxf(fmaxf(s[0][i], s[1][i]), fmaxf(s[2][i], s[3][i]));
      for (int msk = 1; msk < 16; msk <<= 1) mx = fmaxf(mx, __shfl_xor(mx, msk, 16));
      const float mnew = fmaxf(mrow[i], mx);
      const float a = __expf(mrow[i] - mnew);
      float rs = 0.f;
      for (int t = 0; t < 4; ++t) {
        const float p = __expf(s[t][i] - mnew);
        s[t][i] = p;
        rs += p;
      }
      for (int msk = 1; msk < 16; msk <<= 1) rs += __shfl_xor(rs, msk, 16);
      lrow[i] = lrow[i] * a + rs;
      mrow[i] = mnew;
      alpha[i] = a;
    }
    for (int t = 0; t < 4; ++t)
      for (int i = 0; i < 8; ++i) o[t][i] *= alpha[i];

    // stage P wave-privately (DS ops are in-order per wave: no barrier)
    for (int t = 0; t < 4; ++t)
      for (int i = 0; i < 8; ++i)
        Ps[wv][hi * 8 + i][t * 16 + row16] = s[t][i];

    // O += P @ V
    for (int kk = 0; kk < 16; ++kk) {
      const int kd = kk * 4 + hi * 2;
      v2f a;
      a.x = Ps[wv][row16][kd + 0];
      a.y = Ps[wv][row16][kd + 1];
      for (int t = 0; t < 4; ++t) {
        v2f bf;
        bf.x = Vs[buf][kd + 0][t * 16 + row16];
        bf.y = Vs[buf][kd + 1][t * 16 + row16];
        o[t] = WMMA_F32(a, bf, o[t]);
      }
    }

#ifdef HAVE_ASYNC_LDS
    async_wait0();
#endif
    __syncthreads();
    buf ^= 1;
  }

  // normalize + write merged-heads output [B, S, E]
  const int b = bh / Hh, h = bh % Hh;
  for (int t = 0; t < 4; ++t) {
    const int d = t * 16 + row16;
    for (int i = 0; i < 8; ++i) {
      const int srow = q0 + wv * 16 + hi * 8 + i;
      Ob[((size_t)b * Sq + srow) * Edim + h * Dh + d] = o[t][i] / lrow[i];
    }
  }
}

// ---------------------------------------------------------------------------
extern "C" void kernel_launch(void* const* d_in, const int* in_sizes, int n_in,
                              void* d_out, int out_size, void* d_ws,
                              size_t ws_size, hipStream_t stream) {
  const float* hs = (const float*)d_in[0];        // [B,S,E]
  const float* c_attn_w = (const float*)d_in[1];  // [E,3E]
  const float* c_attn_b = (const float*)d_in[2];  // [3E]
  const float* c_proj_w = (const float*)d_in[3];  // [E,E]
  const float* c_proj_b = (const float*)d_in[4];  // [E]
  float* out = (float*)d_out;                     // [B,S,E]

  float* ws = (float*)d_ws;
  const size_t HSZ = (size_t)Bsz * Hh * Sq * Dh;  // 8,388,608 floats
  float* Qw = ws;
  float* Kw = ws + HSZ;
  float* Vw = ws + 2 * HSZ;
  float* Aw = ws + 3 * HSZ;

  gemm_kernel<3 * Edim, true><<<dim3(3 * Edim / 128, (Bsz * Sq) / 128), 256, 0,
                                stream>>>(hs, c_attn_w, c_attn_b, Qw, Kw, Vw);
  attn_kernel<<<dim3(Sq / 64, Bsz * Hh), 128, 0, stream>>>(Qw, Kw, Vw, Aw);
  gemm_kernel<Edim, false><<<dim3(Edim / 128, (Bsz * Sq) / 128), 256, 0,
                             stream>>>(Aw, c_proj_w, c_proj_b, out, nullptr,
                                       nullptr);
}